// BandSplitModule_55662776156418
// MI455X (gfx1250) — compile-verified
//
#include <hip/hip_runtime.h>

typedef float v2f __attribute__((ext_vector_type(2)));
typedef float v4f __attribute__((ext_vector_type(4)));
typedef float v8f __attribute__((ext_vector_type(8)));

constexpr int kNB   = 6;
constexpr int kNSG  = 41;      // total subbands across all bands
constexpr int kT    = 512;
constexpr int kFC   = 128;
constexpr int kCTOT = 22050;
constexpr float kEPS = 1e-5f;

__constant__ int c_SUBP[kNB + 1] = {0, 10, 22, 30, 38, 40, 41};
__constant__ int c_BST[kNB]      = {0, 1000, 4000, 8000, 16000, 20000};
__constant__ int c_WID[kNB]      = {100, 250, 500, 1000, 2000, 2050};

// workspace float offsets
constexpr int WS_PART = 0;                        // 82 groups * 32 chunks * 2
constexpr int WS_MUSD = 82 * 32 * 2;              // 82 * 2 (mu, rstd)
constexpr int WS_S1   = WS_MUSD + 82 * 2;         // 41*128
constexpr int WS_OFF  = WS_S1 + kNSG * kFC;       // 41*128

struct BandPtrs {
  const float* gamma[kNB];
  const float* beta[kNB];
  const float* W[kNB];
  const float* bias[kNB];
};

__device__ __forceinline__ int find_band(int sg) {
  int i = 0;
#pragma unroll
  for (int k = 0; k < kNB - 1; ++k)
    if (sg >= c_SUBP[k + 1]) i = k + 1;
  return i;
}

// ---------------- Kernel 1: per-group partial sums (sum, sumsq) ----------------
__global__ void bs_stats_partial(const float* __restrict__ x, float* __restrict__ ws) {
  const int chunk = blockIdx.x;            // 0..31
  const int g     = blockIdx.y;            // 0..81  (b*41 + sg)
  const int b  = g / kNSG;
  const int sg = g - b * kNSG;
  const int i  = find_band(sg);
  const int w  = c_WID[i];
  const int cBase = c_BST[i] + (sg - c_SUBP[i]) * w;
  const size_t base = ((size_t)b * kCTOT + (size_t)cBase) * (size_t)(kT * 2);
  const int Ng = w * kT * 2;

  float s = 0.f, ss = 0.f;
  for (int e = chunk * 256 + threadIdx.x; e < Ng; e += 32 * 256) {
    float v = x[base + e];
    s += v;
    ss += v * v;
  }

  __shared__ float ls[256];
  __shared__ float lss[256];
  ls[threadIdx.x] = s;
  lss[threadIdx.x] = ss;
  __syncthreads();
  for (int st = 128; st > 0; st >>= 1) {
    if ((int)threadIdx.x < st) {
      ls[threadIdx.x]  += ls[threadIdx.x + st];
      lss[threadIdx.x] += lss[threadIdx.x + st];
    }
    __syncthreads();
  }
  if (threadIdx.x == 0) {
    ws[WS_PART + (g * 32 + chunk) * 2]     = ls[0];
    ws[WS_PART + (g * 32 + chunk) * 2 + 1] = lss[0];
  }
}

// ---------------- Kernel 2: finalize mu / rstd per group ----------------
__global__ void bs_stats_final(float* __restrict__ ws) {
  const int g = blockIdx.x;            // 0..81
  const int lane = threadIdx.x;        // 32 threads
  const int sg = g % kNSG;
  const int i = find_band(sg);
  const float Ng = (float)(c_WID[i] * kT * 2);

  float s  = ws[WS_PART + (g * 32 + lane) * 2];
  float ss = ws[WS_PART + (g * 32 + lane) * 2 + 1];
#pragma unroll
  for (int off = 16; off > 0; off >>= 1) {
    s  += __shfl_xor(s, off, 32);
    ss += __shfl_xor(ss, off, 32);
  }
  if (lane == 0) {
    float mu  = s / Ng;
    float var = ss / Ng - mu * mu;
    ws[WS_MUSD + g * 2]     = mu;
    ws[WS_MUSD + g * 2 + 1] = rsqrtf(var + kEPS);
  }
}

// ---------------- Kernel 3: epilogue constants S1 = sum(gamma*W), OFF = sum(beta*W)+bias ----
__global__ void bs_prep(BandPtrs P, float* __restrict__ ws) {
  const int sg = blockIdx.x;       // 0..40
  const int f  = threadIdx.x;      // 0..127
  const int i  = find_band(sg);
  const int w  = c_WID[i];
  const int sL = sg - c_SUBP[i];
  const float* __restrict__ Wr = P.W[i] + (size_t)(sL * kFC + f) * (size_t)(2 * w);
  const float* __restrict__ ga = P.gamma[i] + sL * w;
  const float* __restrict__ be = P.beta[i]  + sL * w;
  float s1 = 0.f, s2 = 0.f;
  for (int j = 0; j < w; ++j) {
    float wsum = Wr[2 * j] + Wr[2 * j + 1];
    s1 += ga[j] * wsum;
    s2 += be[j] * wsum;
  }
  ws[WS_S1 + sg * kFC + f]  = s1;
  ws[WS_OFF + sg * kFC + f] = s2 + P.bias[i][sL * kFC + f];
}

// ---------------- Kernel 4: WMMA GEMM + fused groupnorm epilogue ----------------
// grid.x = 41 * 2 * 32 ; block = 256 (8 waves). Wave m -> f tile [16m,16m+16),
// block n-tile -> t tile. D = A(W*gamma, 16xK) x B(x, Kx16), fp32 16x16x4 WMMA.
//
// K macro-step of 8 with a per-lane-half permutation of the contraction order:
//   wmma#0 covers actual K = kk + 4*hi + j ; wmma#1 covers K = kk + 2 + 4*hi + j.
// This makes each lane's A data for TWO wmmas one contiguous float4 (b128 load),
// and the two gamma scalars one contiguous float2.
__global__ void __launch_bounds__(256) bs_gemm(const float* __restrict__ x, BandPtrs P,
                                               const float* __restrict__ ws,
                                               float* __restrict__ out) {
  const int blk = blockIdx.x;
  const int nt = blk & 31;           // t tile
  const int b  = (blk >> 5) & 1;     // batch
  const int sg = blk >> 6;           // subband 0..40
  const int i  = find_band(sg);
  const int w  = c_WID[i];
  const int sL = sg - c_SUBP[i];
  const int cBase = c_BST[i] + sL * w;

  const int wave = threadIdx.x >> 5;
  const int lane = threadIdx.x & 31;
  const int hi = lane >> 4;
  const int lo = lane & 15;
  const int fbase = wave * 16;
  const int tbase = nt * 16;
  const int K2 = 2 * w;              // contraction length (always multiple of 4)

  const float* __restrict__ Arow = P.W[i] + (size_t)(sL * kFC + fbase + lo) * (size_t)K2;
  const float* __restrict__ ga   = P.gamma[i] + sL * w;
  const float* __restrict__ Bcol =
      x + ((size_t)b * kCTOT + (size_t)cBase) * (size_t)(kT * 2) + (size_t)(tbase + lo) * 2;

  v8f acc = {0.f, 0.f, 0.f, 0.f, 0.f, 0.f, 0.f, 0.f};

  const int K2m = K2 & ~7;           // macro-step body
#pragma unroll 2
  for (int kk = 0; kk < K2m; kk += 8) {
    const int widx0 = (kk >> 1) + 2 * hi;                  // widx for wmma#0 (even)
    const v4f a4 = *(const v4f*)(Arow + kk + 4 * hi);       // 16B: K = kk+4hi .. +3
    const v2f g2 = *(const v2f*)(ga + widx0);               // gamma[widx0], gamma[widx0+1]
    const v2f b0 = *(const v2f*)(Bcol + (size_t)widx0 * (size_t)(kT * 2));
    const v2f b1 = *(const v2f*)(Bcol + (size_t)(widx0 + 1) * (size_t)(kT * 2));
    v2f a0; a0.x = a4.x * g2.x; a0.y = a4.y * g2.x;
    v2f a1; a1.x = a4.z * g2.y; a1.y = a4.w * g2.y;
    acc = __builtin_amdgcn_wmma_f32_16x16x4_f32(false, a0, false, b0, (short)0, acc,
                                                false, false);
    acc = __builtin_amdgcn_wmma_f32_16x16x4_f32(false, a1, false, b1, (short)0, acc,
                                                false, false);
  }
  if (K2 & 7) {                      // tail of 4 (bands with w=250 or w=2050)
    const int colA = K2m + 2 * hi;
    const int widx = colA >> 1;
    v2f a = *(const v2f*)(Arow + colA);
    const float g = ga[widx];
    a.x *= g;
    a.y *= g;
    const v2f bb = *(const v2f*)(Bcol + (size_t)widx * (size_t)(kT * 2));
    acc = __builtin_amdgcn_wmma_f32_16x16x4_f32(false, a, false, bb, (short)0, acc,
                                                false, false);
  }

  const float mu   = ws[WS_MUSD + (b * kNSG + sg) * 2];
  const float rstd = ws[WS_MUSD + (b * kNSG + sg) * 2 + 1];
  const float negrm = -rstd * mu;

#pragma unroll
  for (int r = 0; r < 8; ++r) {
    const int f = fbase + r + 8 * hi;                  // C/D layout: VGPR r, lane half
    const float s1  = ws[WS_S1 + sg * kFC + f];
    const float ofv = ws[WS_OFF + sg * kFC + f];
    const float y = rstd * acc[r] + negrm * s1 + ofv;
    out[(((size_t)b * kFC + f) * kNSG + sg) * kT + tbase + lo] = y;
  }
}

extern "C" void kernel_launch(void* const* d_in, const int* in_sizes, int n_in,
                              void* d_out, int out_size, void* d_ws, size_t ws_size,
                              hipStream_t stream) {
  (void)in_sizes; (void)n_in; (void)out_size; (void)ws_size;
  const float* x = (const float*)d_in[0];
  BandPtrs P;
  for (int i = 0; i < kNB; ++i) {
    P.gamma[i] = (const float*)d_in[1 + i];        // gn_gamma tuple
    P.beta[i]  = (const float*)d_in[7 + i];        // gn_beta tuple
    P.W[i]     = (const float*)d_in[13 + i];       // conv_w tuple
    P.bias[i]  = (const float*)d_in[19 + i];       // conv_b tuple
  }
  float* ws  = (float*)d_ws;
  float* out = (float*)d_out;

  bs_stats_partial<<<dim3(32, 82), 256, 0, stream>>>(x, ws);
  bs_stats_final<<<82, 32, 0, stream>>>(ws);
  bs_prep<<<kNSG, kFC, 0, stream>>>(P, ws);
  bs_gemm<<<kNSG * 2 * 32, 256, 0, stream>>>(x, P, ws, out);
}